// CrossCaseCL_27685359190208
// MI455X (gfx1250) — compile-verified
//
#include <hip/hip_runtime.h>
#include <stdint.h>

#define BATCH 64
#define SEQ   512
#define DIM   768
#define NT    256          // threads per block (8 wave32)
#define NW    8            // waves per block

typedef __attribute__((ext_vector_type(2))) float v2f;
typedef __attribute__((ext_vector_type(8))) float v8f;

// ---------------------------------------------------------------------------
// Block-wide masked softmax over logit[start .. start+count) stored in LDS.
// Overwrites the region with normalized weights.
// ---------------------------------------------------------------------------
__device__ __forceinline__ void block_softmax(float* logit, int start, int count,
                                              float* red, float* bc, int tid) {
  // --- max ---
  float m = -3.402823466e38f;
  for (int i = tid; i < count; i += NT) m = fmaxf(m, logit[start + i]);
#pragma unroll
  for (int off = 16; off > 0; off >>= 1) m = fmaxf(m, __shfl_xor(m, off, 32));
  if ((tid & 31) == 0) red[tid >> 5] = m;
  __syncthreads();
  if (tid == 0) {
    float mm = red[0];
#pragma unroll
    for (int w = 1; w < NW; ++w) mm = fmaxf(mm, red[w]);
    *bc = mm;
  }
  __syncthreads();
  const float gmax = *bc;

  // --- sum of exp ---
  float s = 0.0f;
  for (int i = tid; i < count; i += NT) {
    float e = expf(logit[start + i] - gmax);
    logit[start + i] = e;
    s += e;
  }
#pragma unroll
  for (int off = 16; off > 0; off >>= 1) s += __shfl_xor(s, off, 32);
  if ((tid & 31) == 0) red[tid >> 5] = s;
  __syncthreads();
  if (tid == 0) {
    float ss = 0.0f;
#pragma unroll
    for (int w = 0; w < NW; ++w) ss += red[w];
    *bc = 1.0f / ss;
  }
  __syncthreads();
  const float inv = *bc;
  for (int i = tid; i < count; i += NT) logit[start + i] *= inv;
}

// ---------------------------------------------------------------------------
// logits[start+r] = scale * dot(emb_row[start+r], vec)  for r in [0, count)
// via V_WMMA_F32_16X16X4_F32. One 16-row tile per wave; B-matrix holds `vec`
// replicated across all 16 columns, so every output column equals the matvec.
// Four independent accumulators break the WMMA->WMMA RAW chain so the four
// WMMAs per K=16 step issue back-to-back behind the clause'd global loads.
//
// ISA layouts (wave32):
//   A (16x4 f32): lane m (0..15)/m+16 holds row m; VGPR0/1 = K pair at 2*hi.
//   B (4x16 f32): VGPR0/1 = rows {2*hi, 2*hi+1}, col = lane&15 (all equal here).
//   C/D (16x16 f32): VGPR r, lanes 0-15 -> M=r, lanes 16-31 -> M=r+8.
// ---------------------------------------------------------------------------
__device__ __forceinline__ void wmma_matvec_logits(const float* __restrict__ eb,
                                                   const float* __restrict__ vec,
                                                   float* __restrict__ logit,
                                                   int start, int count, float scale,
                                                   int lane, int wave) {
  const int ntiles = (count + 15) >> 4;
  const int hi = lane >> 4;      // lane half: K offset selector
  const int m  = lane & 15;      // row within tile
  const int rmax = start + count - 1;

  for (int t = wave; t < ntiles; t += NW) {
    const int r0 = start + (t << 4);
    int row = r0 + m;
    row = (row > rmax) ? rmax : row;              // clamp (duplicate rows, in-bounds)
    const float* rp = eb + (size_t)row * DIM + (hi << 1);
    const float* vp = vec + (hi << 1);

    v8f c0 = {0.f,0.f,0.f,0.f,0.f,0.f,0.f,0.f};
    v8f c1 = c0, c2 = c0, c3 = c0;
    for (int k = 0; k < DIM; k += 16) {           // 48 iterations
      v2f a0 = *(const v2f*)(rp + k);
      v2f a1 = *(const v2f*)(rp + k + 4);
      v2f a2 = *(const v2f*)(rp + k + 8);
      v2f a3 = *(const v2f*)(rp + k + 12);
      v2f b0 = *(const v2f*)(vp + k);
      v2f b1 = *(const v2f*)(vp + k + 4);
      v2f b2 = *(const v2f*)(vp + k + 8);
      v2f b3 = *(const v2f*)(vp + k + 12);
      c0 = __builtin_amdgcn_wmma_f32_16x16x4_f32(false, a0, false, b0, (short)0, c0, false, false);
      c1 = __builtin_amdgcn_wmma_f32_16x16x4_f32(false, a1, false, b1, (short)0, c1, false, false);
      c2 = __builtin_amdgcn_wmma_f32_16x16x4_f32(false, a2, false, b2, (short)0, c2, false, false);
      c3 = __builtin_amdgcn_wmma_f32_16x16x4_f32(false, a3, false, b3, (short)0, c3, false, false);
    }
    v8f c = (c0 + c1) + (c2 + c3);

    if (m == 0) {                                 // lanes 0 and 16 each own 8 rows
      const int rows = count - (t << 4);
#pragma unroll
      for (int r = 0; r < 8; ++r) {
        const int mm = r + (hi << 3);
        if (mm < 16 && mm < rows) logit[r0 + mm] = c[r] * scale;
      }
    }
  }
}

// ---------------------------------------------------------------------------
// Fused per-batch pipeline: one workgroup per batch element.
// ---------------------------------------------------------------------------
__global__ __launch_bounds__(NT) void crosscase_fused(const float* __restrict__ emb,
                                                      const long long* __restrict__ sep,
                                                      float* __restrict__ out) {
  __shared__ float sJ[DIM];       // judgment, later w_r_j
  __shared__ float sLogit[SEQ];   // rj / tj logits -> softmax weights
  __shared__ float sRed[NW];
  __shared__ float sBc;

  const int b    = blockIdx.x;
  const int tid  = threadIdx.x;
  const int lane = tid & 31;
  const int wave = tid >> 5;

  const int s0 = (int)sep[b * 3 + 0];
  const int s1 = (int)sep[b * 3 + 1];
  const int s2 = (int)sep[b * 3 + 2];
  const int F = s0;               // fact rows:   [0, s0)
  const int R = s1 - s0 - 1;      // reason rows: (s0, s1)
  const int J = s2 - s1 - 1;      // judg rows:   (s1, s2)

  const float* eb = emb + (size_t)b * SEQ * DIM;

  // ---- Phase 1: judgment = mean over judg region (column-parallel, coalesced)
  {
    float a0 = 0.f, a1 = 0.f, a2 = 0.f;
    int s = s1 + 1;
    for (; s + 1 < s2; s += 2) {
      const float* r0p = eb + (size_t)s * DIM + tid;
      const float* r1p = r0p + DIM;
      if (s + 6 < s2) __builtin_prefetch(r0p + 6 * DIM, 0, 1);
      a0 += r0p[0]; a1 += r0p[256]; a2 += r0p[512];
      a0 += r1p[0]; a1 += r1p[256]; a2 += r1p[512];
    }
    if (s < s2) {
      const float* r0p = eb + (size_t)s * DIM + tid;
      a0 += r0p[0]; a1 += r0p[256]; a2 += r0p[512];
    }
    const float invJ = 1.0f / (float)J;
    sJ[tid] = a0 * invJ; sJ[tid + 256] = a1 * invJ; sJ[tid + 512] = a2 * invJ;
  }
  __syncthreads();

  // ---- Phase 2: rj logits over reason region (WMMA matvec)
  wmma_matvec_logits(eb, sJ, sLogit, s0 + 1, R, rsqrtf((float)R), lane, wave);
  __syncthreads();

  // ---- Phase 3: softmax over reason region
  block_softmax(sLogit, s0 + 1, R, sRed, &sBc, tid);
  __syncthreads();

  // ---- Phase 4: w_r_j = sum_s a_r[s] * emb[s,:]  (reason region, L2-hot)
  {
    float a0 = 0.f, a1 = 0.f, a2 = 0.f;
    int s = s0 + 1;
    for (; s + 1 < s1; s += 2) {
      const float w0 = sLogit[s], w1 = sLogit[s + 1];
      const float* r0p = eb + (size_t)s * DIM + tid;
      const float* r1p = r0p + DIM;
      a0 += w0 * r0p[0]; a1 += w0 * r0p[256]; a2 += w0 * r0p[512];
      a0 += w1 * r1p[0]; a1 += w1 * r1p[256]; a2 += w1 * r1p[512];
    }
    if (s < s1) {
      const float w0 = sLogit[s];
      const float* r0p = eb + (size_t)s * DIM + tid;
      a0 += w0 * r0p[0]; a1 += w0 * r0p[256]; a2 += w0 * r0p[512];
    }
    sJ[tid] = a0; sJ[tid + 256] = a1; sJ[tid + 512] = a2;   // overwrite judgment
  }
  __syncthreads();

  // ---- Phase 5: tj logits over fact region (WMMA matvec)
  wmma_matvec_logits(eb, sJ, sLogit, 0, F, rsqrtf((float)F), lane, wave);
  __syncthreads();

  // ---- Phase 6: softmax over fact region
  block_softmax(sLogit, 0, F, sRed, &sBc, tid);
  __syncthreads();

  // ---- Phase 7: out = sum_s a_f[s] * emb[s,:] ; plus emb[:,0,:] copy
  {
    float a0 = 0.f, a1 = 0.f, a2 = 0.f;
    int s = 0;
    for (; s + 1 < F; s += 2) {
      const float w0 = sLogit[s], w1 = sLogit[s + 1];
      const float* r0p = eb + (size_t)s * DIM + tid;
      const float* r1p = r0p + DIM;
      a0 += w0 * r0p[0]; a1 += w0 * r0p[256]; a2 += w0 * r0p[512];
      a0 += w1 * r1p[0]; a1 += w1 * r1p[256]; a2 += w1 * r1p[512];
    }
    if (s < F) {
      const float w0 = sLogit[s];
      const float* r0p = eb + (size_t)s * DIM + tid;
      a0 += w0 * r0p[0]; a1 += w0 * r0p[256]; a2 += w0 * r0p[512];
    }
    float* ob = out + (size_t)b * DIM;
    ob[tid] = a0; ob[tid + 256] = a1; ob[tid + 512] = a2;

    float* fb = out + (size_t)BATCH * DIM + (size_t)b * DIM;
    fb[tid]       = eb[tid];
    fb[tid + 256] = eb[tid + 256];
    fb[tid + 512] = eb[tid + 512];
  }
}

extern "C" void kernel_launch(void* const* d_in, const int* in_sizes, int n_in,
                              void* d_out, int out_size, void* d_ws, size_t ws_size,
                              hipStream_t stream) {
  const float*     emb = (const float*)d_in[0];
  const long long* sep = (const long long*)d_in[1];   // reference casts to int64
  float*           out = (float*)d_out;               // [out(B,D) | emb_fact(B,D)]
  (void)in_sizes; (void)n_in; (void)out_size; (void)d_ws; (void)ws_size;

  crosscase_fused<<<dim3(BATCH), dim3(NT), 0, stream>>>(emb, sep, out);
}